// HMPLayer_46497315946645
// MI455X (gfx1250) — compile-verified
//
#include <hip/hip_runtime.h>
#include <hip/hip_bf16.h>

// Problem constants (fixed by the reference)
#define N_NODES 50000
#define N_EDGES 800000
#define H_DIM   128
#define S_DIM   64
#define HID_DIM 128
#define K_TOP   500
#define KP_PAD  512

typedef __attribute__((ext_vector_type(2))) float v2f;
typedef __attribute__((ext_vector_type(8))) float v8f;

// Fast reciprocal (v_rcp_f32, ~1 ULP) — avoids the ~10-op IEEE div expansion
__device__ __forceinline__ float frcp(float x) { return __builtin_amdgcn_rcpf(x); }
__device__ __forceinline__ float silu_f(float x) { return x * frcp(1.0f + __expf(-x)); }

enum { EPI_NONE = 0, EPI_SILU = 1, EPI_RES = 2, EPI_ROWB = 3, EPI_GATE = 4 };

// ---------------------------------------------------------------------------
// Generic fp32 WMMA GEMM: C(M x Ncols) = concat_k[A0|A1](M x Kd) @ B(Kd x Ncols)
// One 16x16 output tile per wave, K-loop of V_WMMA_F32_16X16X4_F32.
// Fragment layout (wave32):
//   A frag (v2f): a[j] = A[tm*16 + (lane&15)][kt*4 + (lane>>4)*2 + j]
//   B frag (v2f): b[j] = B[kt*4 + (lane>>4)*2 + j][tn*16 + (lane&15)]
//   C frag (v8f): c[r] = C[tm*16 + r + 8*(lane>>4)][tn*16 + (lane&15)]
// ---------------------------------------------------------------------------
template <int EPI, bool BTRANS>
__global__ void gemm16(const float* __restrict__ A0, int lda0,
                       const float* __restrict__ A1, int lda1, int k0,
                       int Kd,
                       const float* __restrict__ B, int ldb,
                       const float* __restrict__ bias,
                       float scale,
                       const float* __restrict__ res, int ldres,
                       const float* __restrict__ mul, int ldmul,
                       const float* __restrict__ rowscale,
                       float* __restrict__ C, int ldc,
                       int Mtiles, int Ntiles, int Mvalid, int Nvalid) {
  int wid = blockIdx.x * (blockDim.x >> 5) + (threadIdx.x >> 5);
  if (wid >= Mtiles * Ntiles) return;
  int tm = wid / Ntiles;
  int tn = wid % Ntiles;
  int l  = threadIdx.x & 31;
  int hfw = l >> 4;   // which K-half / M-half this lane covers
  int lm  = l & 15;
  int rowA = tm * 16 + lm;
  int colB = tn * 16 + lm;
  bool aok = rowA < Mvalid;
  bool bok = colB < Nvalid;

  v8f acc = {};
  for (int kt = 0; kt < (Kd >> 2); ++kt) {
    int kb = (kt << 2) + hfw * 2;
    v2f av; av.x = 0.0f; av.y = 0.0f;
    if (aok) {
      if (kb < k0) { const float* p = A0 + (size_t)rowA * lda0 + kb;        av.x = p[0]; av.y = p[1]; }
      else         { const float* p = A1 + (size_t)rowA * lda1 + (kb - k0); av.x = p[0]; av.y = p[1]; }
    }
    v2f bv; bv.x = 0.0f; bv.y = 0.0f;
    if (bok) {
      if (BTRANS) { const float* p = B + (size_t)colB * ldb + kb; bv.x = p[0]; bv.y = p[1]; }
      else        { bv.x = B[(size_t)kb * ldb + colB]; bv.y = B[(size_t)(kb + 1) * ldb + colB]; }
    }
    acc = __builtin_amdgcn_wmma_f32_16x16x4_f32(false, av, false, bv, (short)0, acc, false, false);
  }

#pragma unroll
  for (int r = 0; r < 8; ++r) {
    int cm = tm * 16 + r + hfw * 8;
    int cn = colB;
    if (cm < Mvalid && cn < Nvalid) {
      float v = acc[r] * scale;
      if (EPI == EPI_NONE) {
        if (bias) v += bias[cn];
        C[(size_t)cm * ldc + cn] = v;
      } else if (EPI == EPI_SILU) {
        C[(size_t)cm * ldc + cn] = silu_f(v + bias[cn]);
      } else if (EPI == EPI_RES) {
        C[(size_t)cm * ldc + cn] = res[(size_t)cm * ldres + cn] + v + bias[cn];
      } else if (EPI == EPI_ROWB) {
        C[(size_t)cm * ldc + cn] = v + rowscale[cm] * bias[cn];
      } else if (EPI == EPI_GATE) {
        float g = v + bias[cn];
        C[(size_t)cm * ldc + cn] = res[(size_t)cm * ldres + cn] + g * mul[(size_t)cm * ldmul + cn];
      }
    }
  }
}

template <int EPI, bool BT>
static void run_gemm(hipStream_t st,
                     const float* A0, int lda0, const float* A1, int lda1, int k0, int Kd,
                     const float* B, int ldb, const float* bias, float scale,
                     const float* res, int ldres, const float* mul, int ldmul,
                     const float* rowscale,
                     float* C, int ldc, int M, int Ncols, int Mvalid, int Nvalid) {
  int Mt = (M + 15) / 16, Nt = (Ncols + 15) / 16;
  long long total = (long long)Mt * Nt;
  int blocks = (int)((total + 7) / 8);  // 8 waves (256 threads) per block
  gemm16<EPI, BT><<<blocks, 256, 0, st>>>(A0, lda0, A1, lda1, k0, Kd, B, ldb, bias, scale,
                                          res, ldres, mul, ldmul, rowscale,
                                          C, ldc, Mt, Nt, Mvalid, Nvalid);
}

// ---------------------------------------------------------------------------
// Edge backbone: per wave, one tile of 16 edges.
// A-tile rows are built on the fly: silu(a[src] + bf[dst] + b1), multiplied by
// W2 via 16x16x4 f32 WMMA, epilogue (+b2)*w then atomic segment-sum into agg.
// ---------------------------------------------------------------------------
__global__ void edge_backbone(const float* __restrict__ av, const float* __restrict__ bfv,
                              const float* __restrict__ b1, const float* __restrict__ W2,
                              const float* __restrict__ b2, const int* __restrict__ ei,
                              const float* __restrict__ pos, float* __restrict__ agg, int Etot) {
  int wid = blockIdx.x * (blockDim.x >> 5) + (threadIdx.x >> 5);
  int e0 = wid * 16;
  if (e0 >= Etot) return;
  int l   = threadIdx.x & 31;
  int hfw = l >> 4;
  int lm  = l & 15;
  int e = e0 + lm;
  int s = ei[e];
  int d = ei[Etot + e];
  float dx = pos[s * 3 + 0] - pos[d * 3 + 0];
  float dy = pos[s * 3 + 1] - pos[d * 3 + 1];
  float dz = pos[s * 3 + 2] - pos[d * 3 + 2];
  float wE = frcp(1.0f + dx * dx + dy * dy + dz * dz);
  const float* arow = av  + (size_t)s * H_DIM;
  const float* brow = bfv + (size_t)d * H_DIM;

  v8f acc[8] = {};
  for (int kt = 0; kt < (H_DIM >> 2); ++kt) {
    int kb = (kt << 2) + hfw * 2;
    float x0 = arow[kb] + brow[kb] + b1[kb];
    float x1 = arow[kb + 1] + brow[kb + 1] + b1[kb + 1];
    v2f af; af.x = silu_f(x0); af.y = silu_f(x1);
#pragma unroll
    for (int nt = 0; nt < 8; ++nt) {
      int col = nt * 16 + lm;
      v2f bf2; bf2.x = W2[(size_t)kb * H_DIM + col]; bf2.y = W2[(size_t)(kb + 1) * H_DIM + col];
      acc[nt] = __builtin_amdgcn_wmma_f32_16x16x4_f32(false, af, false, bf2, (short)0, acc[nt], false, false);
    }
  }
#pragma unroll
  for (int r = 0; r < 8; ++r) {
    int m = r + hfw * 8;                 // edge slot within tile
    float wm = __shfl(wE, m, 32);
    int   dm = __shfl(d, m, 32);
#pragma unroll
    for (int nt = 0; nt < 8; ++nt) {
      int n = nt * 16 + lm;
      float val = (acc[nt][r] + b2[n]) * wm;
      atomicAdd(&agg[(size_t)dm * H_DIM + n], val);
    }
  }
}

// ---------------------------------------------------------------------------
// Small helper kernels
// ---------------------------------------------------------------------------
__global__ void zero32(unsigned* p, long long n) {
  long long i = (long long)blockIdx.x * blockDim.x + threadIdx.x;
  if (i < n) p[i] = 0u;
}

// score logit (pre-sigmoid; sigmoid is monotone so top-k is identical) -> sortable key
__global__ void score_keys(const float* __restrict__ t1, const float* __restrict__ Wm2,
                           const float* __restrict__ bm2, unsigned* __restrict__ keys, int n) {
  int row = blockIdx.x * (blockDim.x >> 5) + (threadIdx.x >> 5);
  if (row >= n) return;
  int l = threadIdx.x & 31;
  float s = 0.0f;
  for (int k = l; k < HID_DIM; k += 32) s += t1[(size_t)row * HID_DIM + k] * Wm2[k];
  for (int o = 16; o; o >>= 1) s += __shfl_xor(s, o, 32);
  if (l == 0) {
    float z = s + bm2[0];
    unsigned u = __float_as_uint(z);
    u = (u & 0x80000000u) ? ~u : (u | 0x80000000u);
    keys[row] = u;
  }
}

// Radix-select top-K state: [0]=prefix [1]=kneed [2]=G(count strictly greater) [3]=spare [4]=c0 [5]=c1
__global__ void topk_init(unsigned* hist, int* state) {
  int t = threadIdx.x;
  if (t < 256) hist[t] = 0u;
  if (t == 0) { state[0] = 0; state[1] = K_TOP; state[2] = 0; state[3] = 0; state[4] = 0; state[5] = 0; }
}

__global__ void topk_hist(const unsigned* __restrict__ keys, unsigned* __restrict__ hist,
                          const int* __restrict__ state, int shift, int n) {
  unsigned prefix = (unsigned)state[0];
  unsigned pmask = (shift == 24) ? 0u : (0xFFFFFFFFu << (shift + 8));
  int i = blockIdx.x * blockDim.x + threadIdx.x;
  if (i < n) {
    unsigned k = keys[i];
    if ((k & pmask) == prefix) atomicAdd(&hist[(k >> shift) & 0xFF], 1u);
  }
}

__global__ void topk_scan(unsigned* hist, int* state, int shift) {
  if (threadIdx.x != 0) return;
  int kneed = state[1];
  unsigned prefix = (unsigned)state[0];
  int cum = 0, digit = 0;
  for (int dg = 255; dg >= 0; --dg) {
    int c = (int)hist[dg];
    if (cum + c >= kneed) { digit = dg; break; }
    cum += c;
  }
  state[0] = (int)(prefix | ((unsigned)digit << shift));
  state[1] = kneed - cum;
  if (shift == 0) state[2] = K_TOP - (kneed - cum);
  for (int dg = 0; dg < 256; ++dg) hist[dg] = 0u;
}

__global__ void topk_select(const unsigned* __restrict__ keys, int* state, int* __restrict__ idx, int n) {
  unsigned T = (unsigned)state[0];
  int G = state[2];
  int i = blockIdx.x * blockDim.x + threadIdx.x;
  if (i >= n) return;
  unsigned k = keys[i];
  if (k > T) {
    int p = atomicAdd(&state[4], 1);
    idx[p] = i;
  } else if (k == T) {
    int p = atomicAdd(&state[5], 1);
    if (G + p < K_TOP) idx[G + p] = i;
  }
}

__global__ void scatter_master(const int* __restrict__ idx, int* __restrict__ mask, int* __restrict__ inv) {
  int j = blockIdx.x * blockDim.x + threadIdx.x;
  if (j < K_TOP) { int i = idx[j]; mask[i] = 1; inv[i] = j; }
}

__global__ void gather_masters(const int* __restrict__ idx, const float* __restrict__ hg,
                               const float* __restrict__ pos, float* __restrict__ h_m,
                               float* __restrict__ pos_m) {
  int j = blockIdx.x, t = threadIdx.x;   // grid KP_PAD, block H_DIM
  if (j < K_TOP) {
    int i = idx[j];
    h_m[(size_t)j * H_DIM + t] = hg[(size_t)i * H_DIM + t];
    if (t < 3) pos_m[j * 3 + t] = pos[i * 3 + t];
  } else {
    h_m[(size_t)j * H_DIM + t] = 0.0f;
    if (t < 3) pos_m[j * 3 + t] = 0.0f;
  }
}

__global__ void adj_build(const int* __restrict__ ei, const int* __restrict__ mask,
                          const int* __restrict__ inv, float* __restrict__ adj, int Etot) {
  int e = blockIdx.x * blockDim.x + threadIdx.x;
  if (e >= Etot) return;
  int s = ei[e], d = ei[Etot + e];
  if (mask[s] && mask[d]) adj[(size_t)inv[s] * K_TOP + inv[d]] = 1.0f;   // (adj > 0) -> 1.0
}

__global__ void softmax_rows(float* __restrict__ attn, int n) {
  int row = blockIdx.x * (blockDim.x >> 5) + (threadIdx.x >> 5);
  if (row >= n) return;
  int l = threadIdx.x & 31;
  float mx = -3.4e38f;
  for (int j = l; j < n; j += 32) mx = fmaxf(mx, attn[(size_t)row * n + j]);
  for (int o = 16; o; o >>= 1) mx = fmaxf(mx, __shfl_xor(mx, o, 32));
  float sum = 0.0f;
  for (int j = l; j < n; j += 32) {
    float ev = __expf(attn[(size_t)row * n + j] - mx);
    attn[(size_t)row * n + j] = ev;
    sum += ev;
  }
  for (int o = 16; o; o >>= 1) sum += __shfl_xor(sum, o, 32);
  float is = frcp(sum);
  for (int j = l; j < n; j += 32) attn[(size_t)row * n + j] *= is;
}

__global__ void build_adjw(const float* __restrict__ adj, const float* __restrict__ attn,
                           const float* __restrict__ wvp, float* __restrict__ adjw) {
  int i = blockIdx.x;
  float wv = wvp[0];
  const float thr = 1.0f / (float)K_TOP;
  for (int j = threadIdx.x; j < K_TOP; j += blockDim.x) {
    float bij = 0.5f * attn[(size_t)i * K_TOP + j] + 0.5f * adj[(size_t)i * K_TOP + j];
    float bji = 0.5f * attn[(size_t)j * K_TOP + i] + 0.5f * adj[(size_t)j * K_TOP + i];
    bool aij = (bij > thr) && (i != j);
    bool aji = (bji > thr) && (i != j);
    float v = (aij || aji) ? 1.0f : 0.0f;
    adjw[(size_t)i * K_TOP + j] = adj[(size_t)i * K_TOP + j] + wv * v;
  }
}

// aggm[j][h] = sum_i silu(am[i][h] + bfm[j][h] + b1[h]) * w[i][j];  wsum[j] = sum_i w[i][j]
__global__ void dense_agg(const float* __restrict__ am, const float* __restrict__ bfm,
                          const float* __restrict__ b1, const float* __restrict__ adjw,
                          const float* __restrict__ pos_m, float* __restrict__ aggm,
                          float* __restrict__ wsum) {
  __shared__ float wcol[K_TOP];
  int j = blockIdx.x, t = threadIdx.x;  // grid K_TOP, block H_DIM
  float pjx = pos_m[j * 3 + 0], pjy = pos_m[j * 3 + 1], pjz = pos_m[j * 3 + 2];
  for (int i = t; i < K_TOP; i += blockDim.x) {
    float dx = pos_m[i * 3 + 0] - pjx;
    float dy = pos_m[i * 3 + 1] - pjy;
    float dz = pos_m[i * 3 + 2] - pjz;
    wcol[i] = adjw[(size_t)i * K_TOP + j] * frcp(1.0f + dx * dx + dy * dy + dz * dz);
  }
  __syncthreads();
  float cj = bfm[(size_t)j * H_DIM + t] + b1[t];
  float acc = 0.0f;
  for (int i = 0; i < K_TOP; ++i) acc += silu_f(am[(size_t)i * H_DIM + t] + cj) * wcol[i];
  aggm[(size_t)j * H_DIM + t] = acc;
  if (t == 0) {
    float sm = 0.0f;
    for (int i = 0; i < K_TOP; ++i) sm += wcol[i];
    wsum[j] = sm;
  }
}

__global__ void scatter_hexp(const int* __restrict__ idx, const float* __restrict__ hhier,
                             float* __restrict__ hexp) {
  int j = blockIdx.x, t = threadIdx.x;  // grid K_TOP, block H_DIM
  hexp[(size_t)idx[j] * H_DIM + t] = hhier[(size_t)j * H_DIM + t];
}

// ---------------------------------------------------------------------------
static void zero_launch(hipStream_t st, void* p, long long n) {
  zero32<<<(unsigned)((n + 255) / 256), 256, 0, st>>>((unsigned*)p, n);
}

extern "C" void kernel_launch(void* const* d_in, const int* in_sizes, int n_in,
                              void* d_out, int out_size, void* d_ws, size_t ws_size,
                              hipStream_t stream) {
  (void)in_sizes; (void)n_in; (void)out_size; (void)ws_size;
  const int N = N_NODES, E = N_EDGES, H = H_DIM, S = S_DIM, HID = HID_DIM, K = K_TOP, KP = KP_PAD;

  const float* h        = (const float*)d_in[0];
  const float* h_global = (const float*)d_in[1];
  const float* pos      = (const float*)d_in[2];
  const int*   ei       = (const int*)d_in[3];
  const float* W_a = (const float*)d_in[5];
  const float* W_b = (const float*)d_in[6];
  const float* b1  = (const float*)d_in[7];
  const float* W2  = (const float*)d_in[8];
  const float* b2  = (const float*)d_in[9];
  const float* W3  = (const float*)d_in[10];
  const float* b3  = (const float*)d_in[11];
  const float* w_virtual = (const float*)d_in[12];
  const float* Wm1 = (const float*)d_in[13];
  const float* bm1 = (const float*)d_in[14];
  const float* Wm2 = (const float*)d_in[15];
  const float* bm2 = (const float*)d_in[16];
  const float* Wq  = (const float*)d_in[17];
  const float* Wk  = (const float*)d_in[18];
  const float* Wa1 = (const float*)d_in[19];
  const float* ba1 = (const float*)d_in[20];
  const float* Wa2 = (const float*)d_in[21];
  const float* ba2 = (const float*)d_in[22];
  float* out = (float*)d_out;

  // Workspace layout
  const size_t NH = (size_t)N * H;
  float* bufA = (float*)d_ws;          // a, later hexp
  float* bufB = bufA + NH;             // bf, later g1
  float* bufC = bufB + NH;             // t1, later agg
  unsigned* keys = (unsigned*)(bufC + NH);
  int* idxK  = (int*)(keys + N);
  int* maskN = idxK + K;
  int* invN  = maskN + N;
  unsigned* hist = (unsigned*)(invN + N);
  int* state = (int*)(hist + 256);
  float* adj   = (float*)(state + 8);
  float* adjw  = adj + (size_t)K * K;
  float* attn  = adjw + (size_t)K * K;
  float* h_m   = attn + (size_t)K * K;
  float* pos_m = h_m + (size_t)KP * H;
  float* qb    = pos_m + (size_t)KP * 3;
  float* kb    = qb + (size_t)KP * S;
  float* am    = kb + (size_t)KP * S;
  float* bfm   = am + (size_t)KP * H;
  float* aggm  = bfm + (size_t)KP * H;
  float* wsum  = aggm + (size_t)KP * H;
  float* aggm2 = wsum + KP;
  float* hhier = aggm2 + (size_t)KP * H;

  // ---- Phase A: master scoring + top-K selection --------------------------
  run_gemm<EPI_SILU, false>(stream, h_global, H, nullptr, 0, S, S, Wm1, HID, bm1, 1.0f,
                            nullptr, 0, nullptr, 0, nullptr, bufC, HID, N, HID, N, HID);
  score_keys<<<(N + 7) / 8, 256, 0, stream>>>(bufC, Wm2, bm2, keys, N);
  topk_init<<<1, 256, 0, stream>>>(hist, state);
  for (int shift = 24; shift >= 0; shift -= 8) {
    topk_hist<<<(N + 255) / 256, 256, 0, stream>>>(keys, hist, state, shift, N);
    topk_scan<<<1, 32, 0, stream>>>(hist, state, shift);
  }
  topk_select<<<(N + 255) / 256, 256, 0, stream>>>(keys, state, idxK, N);

  // ---- Phase B: edge backbone -> out[:, 0:H] -------------------------------
  run_gemm<EPI_NONE, false>(stream, h, H, nullptr, 0, H, H, W_a, H, nullptr, 1.0f,
                            nullptr, 0, nullptr, 0, nullptr, bufA, H, N, H, N, H);
  run_gemm<EPI_NONE, false>(stream, h, H, nullptr, 0, H, H, W_b, H, nullptr, 1.0f,
                            nullptr, 0, nullptr, 0, nullptr, bufB, H, N, H, N, H);
  zero_launch(stream, bufC, (long long)NH);  // agg
  {
    int tiles = E / 16;
    edge_backbone<<<(tiles + 7) / 8, 256, 0, stream>>>(bufA, bufB, b1, W2, b2, ei, pos, bufC, E);
  }
  run_gemm<EPI_RES, false>(stream, h, H, bufC, H, H, 2 * H, W3, H, b3, 1.0f,
                           h, H, nullptr, 0, nullptr, out, 2 * H, N, H, N, H);

  // ---- Phase C: master hierarchy ------------------------------------------
  zero_launch(stream, maskN, N);
  zero_launch(stream, invN, N);
  scatter_master<<<(K + 255) / 256, 256, 0, stream>>>(idxK, maskN, invN);
  gather_masters<<<KP, H, 0, stream>>>(idxK, h_global, pos, h_m, pos_m);
  zero_launch(stream, adj, (long long)K * K);
  adj_build<<<(E + 255) / 256, 256, 0, stream>>>(ei, maskN, invN, adj, E);

  run_gemm<EPI_NONE, false>(stream, h_m, H, nullptr, 0, S, S, Wq, S, nullptr, 1.0f,
                            nullptr, 0, nullptr, 0, nullptr, qb, S, KP, S, K, S);
  run_gemm<EPI_NONE, false>(stream, h_m, H, nullptr, 0, S, S, Wk, S, nullptr, 1.0f,
                            nullptr, 0, nullptr, 0, nullptr, kb, S, KP, S, K, S);
  run_gemm<EPI_NONE, true>(stream, qb, S, nullptr, 0, S, S, kb, S, nullptr, 0.125f /* 1/sqrt(S) */,
                           nullptr, 0, nullptr, 0, nullptr, attn, K, KP, KP, K, K);
  softmax_rows<<<(K + 7) / 8, 256, 0, stream>>>(attn, K);
  build_adjw<<<K, 256, 0, stream>>>(adj, attn, w_virtual, adjw);

  run_gemm<EPI_NONE, false>(stream, h_m, H, nullptr, 0, H, H, W_a, H, nullptr, 1.0f,
                            nullptr, 0, nullptr, 0, nullptr, am, H, KP, H, K, H);
  run_gemm<EPI_NONE, false>(stream, h_m, H, nullptr, 0, H, H, W_b, H, nullptr, 1.0f,
                            nullptr, 0, nullptr, 0, nullptr, bfm, H, KP, H, K, H);
  dense_agg<<<K, H, 0, stream>>>(am, bfm, b1, adjw, pos_m, aggm, wsum);
  run_gemm<EPI_ROWB, false>(stream, aggm, H, nullptr, 0, H, H, W2, H, b2, 1.0f,
                            nullptr, 0, nullptr, 0, wsum, aggm2, H, KP, H, K, H);
  run_gemm<EPI_RES, false>(stream, h_m, H, aggm2, H, H, 2 * H, W3, H, b3, 1.0f,
                           h_m, H, nullptr, 0, nullptr, hhier, H, KP, H, K, H);

  zero_launch(stream, bufA, (long long)NH);  // hexp
  scatter_hexp<<<K, H, 0, stream>>>(idxK, hhier, bufA);

  // ---- Phase D: gated global update -> out[:, H:2H] ------------------------
  run_gemm<EPI_SILU, false>(stream, h_global, H, bufA, H, H, 2 * H, Wa1, H, ba1, 1.0f,
                            nullptr, 0, nullptr, 0, nullptr, bufB, H, N, H, N, H);
  run_gemm<EPI_GATE, false>(stream, bufB, H, nullptr, 0, H, H, Wa2, H, ba2, 1.0f,
                            h_global, H, bufA, H, nullptr, out + H, 2 * H, N, H, N, H);
}